// GridSample_1606317769143
// MI455X (gfx1250) — compile-verified
//
#include <hip/hip_runtime.h>

#ifndef __has_builtin
#define __has_builtin(x) 0
#endif

#define L_ 16
#define C_ 16
#define H_ 128
#define W_ 256
#define HW_ (H_ * W_)
#define DECAY_ 0.1f

// The async-LDS builtin takes pointers to 16-byte int vectors:
//   param0: v4i in address space 1 (global), param1: v4i in address space 3 (LDS)
typedef int v4i __attribute__((vector_size(16)));
typedef __attribute__((address_space(1))) v4i* gv4i_p;
typedef __attribute__((address_space(3))) v4i* lv4i_p;

#if __has_builtin(__builtin_amdgcn_global_load_async_to_lds_b128)
#define HAVE_ASYNC_LDS 1
#else
#define HAVE_ASYNC_LDS 0
#endif

// ---------------------------------------------------------------------------
// Kernel 1: transpose img (K,C,H,W) -> imgT (K,H,W,C) so the 16 channels at a
// sample point are one contiguous 64B block (4x float4 gathers per corner).
// ---------------------------------------------------------------------------
__global__ __launch_bounds__(256) void transpose_img_kernel(
    const float* __restrict__ img, float4* __restrict__ imgT4) {
  const int h = blockIdx.x;
  const int k = blockIdx.y;
  const int w = threadIdx.x;

  float v[16];
#pragma unroll
  for (int c = 0; c < C_; ++c)
    v[c] = img[((size_t)(k * C_ + c) * H_ + h) * W_ + w];

  float4* dst = imgT4 + ((size_t)(k * H_ + h) * W_ + w) * 4;
#pragma unroll
  for (int q = 0; q < 4; ++q)
    dst[q] = make_float4(v[4 * q + 0], v[4 * q + 1], v[4 * q + 2], v[4 * q + 3]);
}

// ---------------------------------------------------------------------------
// Kernel 2: one block = one (t, h) output row (256 threads = 256 w's).
// LDS-stage all 32 flow rows (k, {x,y}) for this h via async loads, then for
// each k <= t do the bilinear gather + weighted accumulate into 16 channel
// accumulators.
// ---------------------------------------------------------------------------
template <bool FAST>
__global__ __launch_bounds__(256) void warp_acc_kernel(
    const float* __restrict__ img,     // original layout (slow path only)
    const float4* __restrict__ imgT4,  // transposed layout (fast path only)
    const float* __restrict__ flow,    // (1, L, 2, H, W)
    const float* __restrict__ mask,    // (L, L)
    const float* __restrict__ dist,    // (L, L)
    float* __restrict__ out)           // (1, L, C, H, W)
{
  const int h = blockIdx.x;
  const int t = blockIdx.y;
  const int w = threadIdx.x;

  __shared__ __align__(16) float s_flow[2 * L_ * W_]; // 32 KB
  __shared__ float s_wk[L_];

  // ---- Stage the 32 flow rows (k*2+comp, h, :) into LDS -------------------
#if HAVE_ASYNC_LDS
#pragma unroll
  for (int i = 0; i < 8; ++i) {
    const int f4 = threadIdx.x + i * 256; // float4 index within 32KB tile
    const int j = f4 >> 6;                // which (k,comp) row, 0..31
    const int r = f4 & 63;                // float4 within the 1KB row
    const float* g = flow + ((size_t)j * H_ + h) * W_ + (r << 2);
    float* l = s_flow + j * W_ + (r << 2);
    __builtin_amdgcn_global_load_async_to_lds_b128((gv4i_p)(void*)(const_cast<float*>(g)),
                                                   (lv4i_p)l, 0, 0);
  }
#if __has_builtin(__builtin_amdgcn_s_wait_asynccnt)
  __builtin_amdgcn_s_wait_asynccnt(0);
#else
  asm volatile("s_wait_asynccnt 0" ::: "memory");
#endif
#else
#pragma unroll
  for (int i = 0; i < 8; ++i) {
    const int f4 = threadIdx.x + i * 256;
    const int j = f4 >> 6;
    const int r = f4 & 63;
    const float4* g = (const float4*)(flow + ((size_t)j * H_ + h) * W_) + r;
    *((float4*)(s_flow + j * W_) + r) = *g;
  }
#endif

  if (threadIdx.x < L_) {
    const int k = threadIdx.x;
    s_wk[k] = mask[t * L_ + k] * expf(-DECAY_ * dist[t * L_ + k]);
  }
  __syncthreads();

  const float ftx = s_flow[(2 * t + 0) * W_ + w];
  const float fty = s_flow[(2 * t + 1) * W_ + w];

  float acc[16];
#pragma unroll
  for (int c = 0; c < 16; ++c) acc[c] = 0.0f;

  for (int k = 0; k <= t; ++k) {
    const float fkx = s_flow[(2 * k + 0) * W_ + w];
    const float fky = s_flow[(2 * k + 1) * W_ + w];
    const float wk = s_wk[k];

    // ix = (gx_wrapped+1)*W/2-0.5 simplifies to w + 128*(dfx); wrap == &255.
    const float ix = (float)w + 128.0f * (ftx - fkx);
    const float iy = (float)h + 64.0f * (fty - fky);
    const float x0f = floorf(ix);
    const float y0f = floorf(iy);
    const float wx = ix - x0f;
    const float wy = iy - y0f;
    const int x0 = ((int)x0f) & (W_ - 1);
    const int x1 = (x0 + 1) & (W_ - 1);
    const int y0i = (int)y0f;
    const int y0 = min(max(y0i, 0), H_ - 1);
    const int y1 = min(max(y0i + 1, 0), H_ - 1);

    const float w00 = wk * (1.0f - wx) * (1.0f - wy);
    const float w01 = wk * (1.0f - wx) * wy;
    const float w10 = wk * wx * (1.0f - wy);
    const float w11 = wk * wx * wy;

    if (FAST) {
      const int base = k * HW_;
      const float4* A = imgT4 + (size_t)(base + y0 * W_ + x0) * 4;
      const float4* Bq = imgT4 + (size_t)(base + y1 * W_ + x0) * 4;
      const float4* Cq = imgT4 + (size_t)(base + y0 * W_ + x1) * 4;
      const float4* Dq = imgT4 + (size_t)(base + y1 * W_ + x1) * 4;
      // Speculative prefetch of next source frame at the same spot
      // (flow fields are smooth; next k's corner lands nearby).
      if (k < t) __builtin_prefetch((const void*)(A + (size_t)HW_ * 4), 0, 1);
#pragma unroll
      for (int q = 0; q < 4; ++q) {
        const float4 a = A[q], b = Bq[q], c4 = Cq[q], d = Dq[q];
        acc[4 * q + 0] = fmaf(d.x, w11, fmaf(c4.x, w10, fmaf(b.x, w01, fmaf(a.x, w00, acc[4 * q + 0]))));
        acc[4 * q + 1] = fmaf(d.y, w11, fmaf(c4.y, w10, fmaf(b.y, w01, fmaf(a.y, w00, acc[4 * q + 1]))));
        acc[4 * q + 2] = fmaf(d.z, w11, fmaf(c4.z, w10, fmaf(b.z, w01, fmaf(a.z, w00, acc[4 * q + 2]))));
        acc[4 * q + 3] = fmaf(d.w, w11, fmaf(c4.w, w10, fmaf(b.w, w01, fmaf(a.w, w00, acc[4 * q + 3]))));
      }
    } else {
      const float* bp = img + (size_t)k * C_ * HW_;
      const int iA = y0 * W_ + x0, iB = y1 * W_ + x0;
      const int iC = y0 * W_ + x1, iD = y1 * W_ + x1;
#pragma unroll
      for (int c = 0; c < 16; ++c) {
        const float* pc = bp + (size_t)c * HW_;
        acc[c] = fmaf(pc[iD], w11, fmaf(pc[iC], w10, fmaf(pc[iB], w01, fmaf(pc[iA], w00, acc[c]))));
      }
    }
  }

  // out[((t*C + c)*H + h)*W + w]; coalesced along w per channel.
  float* ob = out + ((size_t)t * C_ * H_ + h) * W_ + w;
#pragma unroll
  for (int c = 0; c < 16; ++c) ob[(size_t)c * HW_] = acc[c];
}

// ---------------------------------------------------------------------------
extern "C" void kernel_launch(void* const* d_in, const int* in_sizes, int n_in,
                              void* d_out, int out_size, void* d_ws, size_t ws_size,
                              hipStream_t stream) {
  const float* img = (const float*)d_in[0];       // (1,16,16,128,256)
  const float* cum_flow = (const float*)d_in[1];  // (1,16,2,128,256)
  const float* mask = (const float*)d_in[2];      // (16,16)
  const float* dist = (const float*)d_in[3];      // (16,16)
  float* out = (float*)d_out;

  const size_t needT = (size_t)L_ * H_ * W_ * C_ * sizeof(float); // 33.5 MB
  dim3 grid(H_, L_);
  dim3 block(256);

  if (ws_size >= needT) {
    float4* imgT4 = (float4*)d_ws;
    transpose_img_kernel<<<dim3(H_, L_), 256, 0, stream>>>(img, imgT4);
    warp_acc_kernel<true><<<grid, block, 0, stream>>>(img, imgT4, cum_flow, mask, dist, out);
  } else {
    warp_acc_kernel<false><<<grid, block, 0, stream>>>(img, nullptr, cum_flow, mask, dist, out);
  }
}